// grounding_model_48430051230463
// MI455X (gfx1250) — compile-verified
//
#include <hip/hip_runtime.h>
#include <stdint.h>

// ---------------- threefry2x32, exact JAX schedule, key = (0, 1234) ----------
__device__ __forceinline__ void threefry_1234(uint32_t x0, uint32_t x1,
                                              uint32_t& o0, uint32_t& o1) {
  const uint32_t ks0 = 0u;
  const uint32_t ks1 = 1234u;
  const uint32_t ks2 = 1234u ^ 0x1BD11BDAu;
  x0 += ks0; x1 += ks1;
#define TF_RND(r) { x0 += x1; x1 = (x1 << (r)) | (x1 >> (32 - (r))); x1 ^= x0; }
  TF_RND(13) TF_RND(15) TF_RND(26) TF_RND(6)   x0 += ks1; x1 += ks2 + 1u;
  TF_RND(17) TF_RND(29) TF_RND(16) TF_RND(24)  x0 += ks2; x1 += ks0 + 2u;
  TF_RND(13) TF_RND(15) TF_RND(26) TF_RND(6)   x0 += ks0; x1 += ks1 + 3u;
  TF_RND(17) TF_RND(29) TF_RND(16) TF_RND(24)  x0 += ks1; x1 += ks2 + 4u;
  TF_RND(13) TF_RND(15) TF_RND(26) TF_RND(6)   x0 += ks2; x1 += ks0 + 5u;
#undef TF_RND
  o0 = x0; o1 = x1;
}

__device__ __forceinline__ void insert5(unsigned long long k,
    unsigned long long& t0, unsigned long long& t1, unsigned long long& t2,
    unsigned long long& t3, unsigned long long& t4) {
  if (k > t4) {
    if (k > t2) {
      if (k > t1) {
        if (k > t0) { t4=t3; t3=t2; t2=t1; t1=t0; t0=k; }
        else        { t4=t3; t3=t2; t2=t1; t1=k; }
      } else        { t4=t3; t3=t2; t2=k; }
    } else {
      if (k > t3)   { t4=t3; t3=k; }
      else          { t4=k; }
    }
  }
}

__device__ __forceinline__ unsigned long long wave_max_u64(unsigned long long m) {
  #pragma unroll
  for (int off = 1; off < 32; off <<= 1) {
    unsigned long long o = __shfl_xor(m, off, 32);
    if (o > m) m = o;
  }
  return m;
}

// ---------------- Kernel 1: vit_feature (16,256,1024) -> vit_corr (1024,16,256)
// Async global->LDS loads with per-lane transposed LDS placement (CDNA5 path).
__global__ __launch_bounds__(256)
void k_transpose_vit(const float* __restrict__ vit, float* __restrict__ out1) {
  __shared__ float tile[32 * 33];
  const int tx = threadIdx.x;       // 0..31  (r within tile on load, d on store)
  const int ty = threadIdx.y;       // 0..7
  const int r0 = blockIdx.x * 32;
  const int d0 = blockIdx.y * 32;
  const int b  = blockIdx.z;

  const unsigned ldsBase = (unsigned)(size_t)(void*)tile;  // LDS byte offset
  #pragma unroll
  for (int tt = 0; tt < 4; ++tt) {
    const int dl = ty + tt * 8;
    const unsigned goff = (unsigned)((((b * 256 + (d0 + dl)) * 1024) + (r0 + tx)) * 4);
    const unsigned loff = ldsBase + (unsigned)((tx * 33 + dl) * 4);  // transposed placement
    asm volatile("global_load_async_to_lds_b32 %0, %1, %2"
                 :: "v"(loff), "v"(goff), "s"(vit) : "memory");
  }
  asm volatile("s_wait_asynccnt 0" ::: "memory");
  __syncthreads();
  #pragma unroll
  for (int tt = 0; tt < 4; ++tt) {
    const int rl = ty + tt * 8;
    out1[(size_t)((r0 + rl) * 16 + b) * 256 + (d0 + tx)] = tile[rl * 33 + tx];
  }
}

// ---------------- Kernel 2: top-4-of-20 + gather -> lag_corr (1024,16,4,256)
__global__ __launch_bounds__(256)
void k_lag_topk(const float* __restrict__ lag_map,   // (16,20,1024)
                const float* __restrict__ lag_feat,  // (16,20,256)
                float* __restrict__ out2) {          // (1024,16,4,256)
  const int tid  = blockIdx.x * 256 + threadIdx.x;
  const int wave = tid >> 5;          // 0..16383
  const int lane = tid & 31;
  const int b = wave & 15;
  const int r = wave >> 4;

  unsigned long long key = 0ull;
  if (lane < 20) {
    const unsigned bi = __float_as_uint(lag_map[((size_t)b * 20 + lane) * 1024 + r]);
    const unsigned u  = (bi & 0x80000000u) ? ~bi : (bi | 0x80000000u); // order-preserving
    key = (((unsigned long long)u) << 5) | (unsigned)(31 - lane);      // stable tie-break
  }
  int cols[4];
  #pragma unroll
  for (int k = 0; k < 4; ++k) {
    const unsigned long long m = wave_max_u64(key);
    cols[k] = 31 - (int)(m & 31ull);
    if (key == m) key = 0ull;         // keys are unique per lane
  }
  const float4* __restrict__ src = (const float4*)lag_feat;
  float4*       __restrict__ dst = (float4*)out2;
  const size_t dbase = (size_t)(r * 16 + b) * 4 * 64;   // rows of 64 float4
  #pragma unroll
  for (int k = 0; k < 4; ++k) {
    const size_t s = ((size_t)b * 20 + cols[k]) * 64;
    const size_t d = dbase + (size_t)k * 64;
    dst[d + lane]      = src[s + lane];
    dst[d + lane + 32] = src[s + lane + 32];
  }
}

// ---------------- Kernel 3: threefry uniforms + top-5 + gather -> neg (1024,16,5,256)
// One wave handles flat rows p and p+8192 (they share threefry counter pairs).
__global__ __launch_bounds__(256)
void k_neg(const float* __restrict__ out1,   // (1024,16,256) = vit_corr (already written)
           float* __restrict__ out3) {       // (1024,16,5,256)
  const int tid  = blockIdx.x * 256 + threadIdx.x;
  const unsigned p    = (unsigned)(tid >> 5);   // 0..8191
  const int      lane = tid & 31;

  const int rA = (int)(p >> 4), bA = (int)(p & 15u);
  const unsigned pB = p + 8192u;
  const int rB = (int)(pB >> 4), bB = (int)(pB & 15u);
  const unsigned base = p * 1024u;              // counter base; half-offset H = 8388608

  unsigned long long a0=0,a1=0,a2=0,a3=0,a4=0;
  unsigned long long c0=0,c1=0,c2=0,c3=0,c4=0;

  #pragma unroll 2
  for (int t = 0; t < 32; ++t) {
    const unsigned j = (unsigned)lane * 32u + (unsigned)t;
    const unsigned i = base + j;
    uint32_t o0, o1;
    threefry_1234(i, i + 8388608u, o0, o1);
    unsigned long long kA = ((((unsigned long long)(o0 >> 9)) << 10) | (1023u - j)) + 1ull;
    if (bA == 15 && j == (unsigned)rA) kA = 0ull;      // diagonal exclusion (-1.0)
    insert5(kA, a0, a1, a2, a3, a4);
    unsigned long long kB = ((((unsigned long long)(o1 >> 9)) << 10) | (1023u - j)) + 1ull;
    if (bB == 15 && j == (unsigned)rB) kB = 0ull;
    insert5(kB, c0, c1, c2, c3, c4);
  }

  int idxA[5], idxB[5];
  #pragma unroll
  for (int n = 0; n < 5; ++n) {
    const unsigned long long m = wave_max_u64(a0);
    idxA[n] = 1023 - (int)((m - 1ull) & 1023ull);
    if (a0 == m) { a0=a1; a1=a2; a2=a3; a3=a4; a4=0ull; }
  }
  #pragma unroll
  for (int n = 0; n < 5; ++n) {
    const unsigned long long m = wave_max_u64(c0);
    idxB[n] = 1023 - (int)((m - 1ull) & 1023ull);
    if (c0 == m) { c0=c1; c1=c2; c2=c3; c3=c4; c4=0ull; }
  }

  const float4* __restrict__ src = (const float4*)out1;
  float4*       __restrict__ dst = (float4*)out3;
  #pragma unroll
  for (int n = 0; n < 5; ++n) {   // row A
    const size_t s = ((size_t)idxA[n] * 16 + 15) * 64;   // vit_corr[idx, 15, :]
    const size_t d = ((size_t)p * 5 + n) * 64;
    dst[d + lane]      = src[s + lane];
    dst[d + lane + 32] = src[s + lane + 32];
  }
  #pragma unroll
  for (int n = 0; n < 5; ++n) {   // row B
    const size_t s = ((size_t)idxB[n] * 16 + 15) * 64;
    const size_t d = ((size_t)pB * 5 + n) * 64;
    dst[d + lane]      = src[s + lane];
    dst[d + lane + 32] = src[s + lane + 32];
  }
}

extern "C" void kernel_launch(void* const* d_in, const int* in_sizes, int n_in,
                              void* d_out, int out_size, void* d_ws, size_t ws_size,
                              hipStream_t stream) {
  (void)in_sizes; (void)n_in; (void)out_size; (void)d_ws; (void)ws_size;
  const float* vit_feature = (const float*)d_in[0];   // (16,256,1024)
  const float* lag_feature = (const float*)d_in[1];   // (16,20,256)
  const float* lag_vit_map = (const float*)d_in[2];   // (16,20,1024)

  float* out1 = (float*)d_out;                 // vit_corr: 1024*16*256   = 4,194,304
  float* out2 = out1 + (size_t)4194304;        // lag_corr: 1024*16*4*256 = 16,777,216
  float* out3 = out2 + (size_t)16777216;       // neg:      1024*16*5*256 = 20,971,520

  dim3 g1(32, 8, 16), b1(32, 8);
  k_transpose_vit<<<g1, b1, 0, stream>>>(vit_feature, out1);
  k_lag_topk<<<2048, 256, 0, stream>>>(lag_vit_map, lag_feature, out2);
  k_neg<<<1024, 256, 0, stream>>>(out1, out3);
}